// EnhancedMambaLayer_84920093377122
// MI455X (gfx1250) — compile-verified
//
#include <hip/hip_runtime.h>
#include <hip/hip_bf16.h>
#include <math.h>
#include <stdint.h>

// ---------------------------------------------------------------------------
// EnhancedMambaLayer for MI455X (gfx1250, wave32, WMMA).
// f32 end-to-end (matches float32 reference). GEMMs: V_WMMA_F32_16X16X4_F32
// with the W slab staged in LDS (double-buffered async global->LDS DMA,
// tracked with ASYNCcnt). Selective scan: one lane per channel, 16-state
// recurrence in VGPRs, B/C staged through double-buffered LDS.
// ---------------------------------------------------------------------------

#define D_MODEL 512
#define D_INNER 1024
#define D_STATE 16
#define DT_RANK 32
#define BATCH   2
#define SEQ     2048
#define NTOK    (BATCH * SEQ)   // 4096 token rows
#define KT      32              // K-chunk staged in LDS per GEMM iteration

typedef float v2f __attribute__((ext_vector_type(2)));
typedef float v4f __attribute__((ext_vector_type(4)));
typedef float v8f __attribute__((ext_vector_type(8)));

// ---------------------------------------------------------------------------
// Async global->LDS copy (gfx1250 GLOBAL_LOAD_ASYNC_TO_LDS_B128, ASYNCcnt).
// Probe round 2 confirmed the builtin exists with signature
//   (v4i addrspace(1)*, v4i addrspace(3)*, i32 offset, i32 cpol)
// where v4i = int __attribute__((vector_size(16))).
// ---------------------------------------------------------------------------
#if defined(__has_builtin)
#if __has_builtin(__builtin_amdgcn_global_load_async_to_lds_b128)
#define HAVE_ASYNC_COPY 1
#endif
#endif
#ifndef HAVE_ASYNC_COPY
#define HAVE_ASYNC_COPY 0
#endif

typedef int v4i_vs __attribute__((vector_size(16)));
typedef __attribute__((address_space(1))) v4i_vs as1_v4i;
typedef __attribute__((address_space(3))) v4i_vs as3_v4i;

__device__ __forceinline__ void async_copy16(const float* g, float* l) {
#if HAVE_ASYNC_COPY
    __builtin_amdgcn_global_load_async_to_lds_b128(
        (as1_v4i*)(uintptr_t)g,
        (as3_v4i*)(uintptr_t)l,
        0, 0);
#else
    *(v4f*)l = *(const v4f*)g;          // global_load_b128 + ds_store_b128
#endif
}

template <int N>
__device__ __forceinline__ void wait_async() {
#if HAVE_ASYNC_COPY
#if __has_builtin(__builtin_amdgcn_s_wait_asynccnt)
    __builtin_amdgcn_s_wait_asynccnt(N);
#else
    asm volatile("s_wait_asynccnt %0" :: "i"(N) : "memory");
#endif
#endif
}

// ---------------------------------------------------------------------------
// Device helpers
// ---------------------------------------------------------------------------
__device__ __forceinline__ float gelu_exact(float x) {
    return 0.5f * x * (1.0f + erff(x * 0.70710678118654752f));
}
__device__ __forceinline__ float softplus_f(float x) {
    return (x > 20.0f) ? x : log1pf(expf(x));
}
__device__ __forceinline__ float sigmoid_f(float x) {
    return 1.0f / (1.0f + expf(-x));
}

// ---------------------------------------------------------------------------
// WMMA f32 GEMM:  C[M,N] = epilogue(A[M,K(lda)] * W[K,N])
// Block = 256 threads = 8 waves stacked along M -> 128x64 block tile.
// Grid = (M/128, N/64). Requires M % 128 == 0, K % KT == 0, N % 64 == 0.
//
// W K-slabs (KT x 64) are staged in LDS (row-major, mirrors global rows so
// 16B async copies land contiguously), double-buffered so the next slab's
// DMA overlaps the WMMA stream. All 8 waves share the slab (8x traffic dedup).
//
// A fragment (16x4 f32): lanes 0-15 -> rows, K=k,k+1; lanes 16-31 -> K=k+2,3.
// B fragment (4x16): VGPR0/1 = K rows kk,kk+1 (lanes 0-15) / kk+2,kk+3
// (lanes 16-31), N striped across lanes. C/D: VGPR i -> row i / 8+i.
// ---------------------------------------------------------------------------
template <int HAS_BIAS, int ACT, int HAS_RES>   // ACT: 0 none, 1 gelu, 2 softplus
__global__ __launch_bounds__(256)
void gemm_wmma_f32(const float* __restrict__ A, const float* __restrict__ W,
                   const float* __restrict__ bias, const float* __restrict__ res,
                   float* __restrict__ C, int M, int N, int K, int lda)
{
    __shared__ __align__(16) float wlds[2][KT * 64];

    const int tid  = threadIdx.x;
    const int lane = tid & 31;
    const int wave = tid >> 5;
    const int m0 = (blockIdx.x * 8 + wave) * 16;
    const int n0 = blockIdx.y * 64;
    const int kh  = lane >> 4;   // 0: K lo pair, 1: K hi pair
    const int l15 = lane & 15;

    // Cooperative slab copy: 512 x 16B ops per chunk, 2 per thread.
    const int cp_k   = tid >> 4;        // 0..15
    const int cp_seg = (tid & 15) * 4;  // float offset within 64-wide row

    const float* __restrict__ arow = A + (size_t)(m0 + l15) * lda;
    const int nchunks = K / KT;

    auto issue = [&](int c) {
        float* dst = wlds[c & 1];
        const float* g = W + (size_t)(c * KT + cp_k) * N + n0 + cp_seg;
        async_copy16(g, dst + cp_k * 64 + cp_seg);
        async_copy16(g + (size_t)16 * N, dst + (cp_k + 16) * 64 + cp_seg);
    };

    v8f acc[4] = {};
    issue(0);

    for (int c = 0; c < nchunks; ++c) {
        if (c + 1 < nchunks) {
            issue(c + 1);
            wait_async<2>();            // current slab done; next still in flight
        } else {
            wait_async<0>();
        }
        __syncthreads();

        const float* __restrict__ buf = wlds[c & 1];
        const int kb = c * KT;
        if (c + 1 < nchunks)
            __builtin_prefetch(arow + kb + KT, 0, 1);   // next A chunk

        #pragma unroll
        for (int kl = 0; kl < KT; kl += 4) {
            const int kk = kl + kh * 2;
            v2f a = *(const v2f*)(arow + kb + kk);       // global_load_b64
            #pragma unroll
            for (int j = 0; j < 4; ++j) {
                const float* bp = buf + kk * 64 + j * 16 + l15;
                v2f b;
                b.x = bp[0];                             // ds_load row kk
                b.y = bp[64];                            // ds_load row kk+1
                acc[j] = __builtin_amdgcn_wmma_f32_16x16x4_f32(
                    false, a, false, b, (short)0, acc[j], false, false);
            }
        }
        __syncthreads();
    }

    // Epilogue + store
    #pragma unroll
    for (int j = 0; j < 4; ++j) {
        const int col = n0 + j * 16 + l15;
        const float bv = HAS_BIAS ? bias[col] : 0.0f;
        #pragma unroll
        for (int i = 0; i < 8; ++i) {
            const int row = m0 + kh * 8 + i;
            float v = acc[j][i];
            if (HAS_BIAS) v += bv;
            if (ACT == 1) v = gelu_exact(v);
            else if (ACT == 2) v = softplus_f(v);
            if (HAS_RES) v += res[(size_t)row * N + col];
            C[(size_t)row * N + col] = v;
        }
    }
}

// ---------------------------------------------------------------------------
// LayerNorm over rows of 512. One wave per row, 8 rows per block.
// ---------------------------------------------------------------------------
__global__ __launch_bounds__(256)
void layernorm_k(const float* __restrict__ x, const float* __restrict__ g,
                 const float* __restrict__ b, float* __restrict__ o, int nrows)
{
    const int lane = threadIdx.x & 31;
    const int wave = threadIdx.x >> 5;
    const int row = blockIdx.x * 8 + wave;
    if (row >= nrows) return;

    const float* __restrict__ xr = x + (size_t)row * D_MODEL;
    float v[16];
    float s = 0.f, s2 = 0.f;
    #pragma unroll
    for (int i = 0; i < 16; ++i) {
        v[i] = xr[lane + 32 * i];
        s += v[i];
        s2 += v[i] * v[i];
    }
    #pragma unroll
    for (int off = 16; off > 0; off >>= 1) {
        s  += __shfl_xor(s,  off, 32);
        s2 += __shfl_xor(s2, off, 32);
    }
    const float mean = s * (1.0f / D_MODEL);
    const float var  = s2 * (1.0f / D_MODEL) - mean * mean;
    const float inv  = rsqrtf(var + 1e-5f);

    float* __restrict__ orow = o + (size_t)row * D_MODEL;
    #pragma unroll
    for (int i = 0; i < 16; ++i) {
        const int c = lane + 32 * i;
        orow[c] = (v[i] - mean) * inv * g[c] + b[c];
    }
}

// ---------------------------------------------------------------------------
// Depthwise causal conv (width 4) + bias + SiLU.
// xi = left half of xz (token row stride 2*D_INNER).  xc contiguous.
// ---------------------------------------------------------------------------
__global__ __launch_bounds__(256)
void conv_silu_k(const float* __restrict__ xz, const float* __restrict__ w,
                 const float* __restrict__ bias, float* __restrict__ xc)
{
    const int idx = blockIdx.x * blockDim.x + threadIdx.x;  // (b*SEQ+t)*D_INNER+d
    if (idx >= NTOK * D_INNER) return;
    const int d  = idx & (D_INNER - 1);
    const int t  = (idx >> 10) & (SEQ - 1);
    const int bz = idx >> 21;

    const float* __restrict__ col = xz + (size_t)bz * SEQ * (2 * D_INNER) + d;
    float acc = bias[d];
    #pragma unroll
    for (int j = 0; j < 4; ++j) {
        const int tt = t - 3 + j;
        if (tt >= 0) acc += w[d * 4 + j] * col[(size_t)tt * (2 * D_INNER)];
    }
    xc[idx] = acc * sigmoid_f(acc);
}

// ---------------------------------------------------------------------------
// Selective scan. One lane per (batch, channel): 16-state recurrence in VGPRs.
// B_t/C_t (16 floats each) staged through double-buffered LDS in 32-step
// chunks; the next chunk's async DMA overlaps the recurrence. Inner-loop LDS
// reads are wave-broadcasts (conflict-free).
// Grid: BATCH * (D_INNER/256) = 8 blocks of 256 threads.
// ---------------------------------------------------------------------------
__global__ __launch_bounds__(256)
void scan_k(const float* __restrict__ dtb, const float* __restrict__ xdbl,
            const float* __restrict__ xc, const float* __restrict__ Alog,
            float* __restrict__ ys)
{
    __shared__ __align__(16) float Bsh[2][32 * D_STATE];
    __shared__ __align__(16) float Csh[2][32 * D_STATE];

    const int tid = threadIdx.x;
    const int bz = blockIdx.x >> 2;                 // batch
    const int d  = (blockIdx.x & 3) * 256 + tid;    // channel

    float negA[D_STATE], h[D_STATE];
    #pragma unroll
    for (int s = 0; s < D_STATE; ++s) {
        negA[s] = -expf(Alog[d * D_STATE + s]);     // A = -exp(Alog)
        h[s] = 0.0f;
    }

    const size_t tokbase = (size_t)bz * SEQ;
    const int NXD = DT_RANK + 2 * D_STATE;          // 64
    const int nchunks = SEQ / 32;

    // One 16B async op per thread per chunk: 32 tokens x (16B B + 16B C) x4.
    const int cp_tt   = tid >> 3;                   // token in chunk, 0..31
    const int cp_part = tid & 7;                    // 0..3 -> B, 4..7 -> C

    auto issue = [&](int c) {
        const int buf = c & 1;
        const size_t rb = (tokbase + (size_t)c * 32 + cp_tt) * NXD;
        if (cp_part < 4)
            async_copy16(&xdbl[rb + DT_RANK + cp_part * 4],
                         &Bsh[buf][cp_tt * D_STATE + cp_part * 4]);
        else
            async_copy16(&xdbl[rb + DT_RANK + D_STATE + (cp_part - 4) * 4],
                         &Csh[buf][cp_tt * D_STATE + (cp_part - 4) * 4]);
    };

    issue(0);
    for (int c = 0; c < nchunks; ++c) {
        if (c + 1 < nchunks) {
            issue(c + 1);
            wait_async<1>();
        } else {
            wait_async<0>();
        }
        __syncthreads();

        const float* __restrict__ Bb = Bsh[c & 1];
        const float* __restrict__ Cb = Csh[c & 1];

        for (int tt = 0; tt < 32; ++tt) {
            const size_t tok = tokbase + (size_t)c * 32 + tt;
            const float dtv = dtb[tok * D_INNER + d];
            const float du  = dtv * xc[tok * D_INNER + d];
            float y = 0.0f;
            #pragma unroll
            for (int s = 0; s < D_STATE; ++s) {
                const float dA = expf(dtv * negA[s]);
                h[s] = dA * h[s] + du * Bb[tt * D_STATE + s];
                y += h[s] * Cb[tt * D_STATE + s];
            }
            ys[tok * D_INNER + d] = y;
        }
        __syncthreads();
    }
}

// ---------------------------------------------------------------------------
// Gating: y = (ys + xc*D) * z * sigmoid(z), z = right half of xz. In-place.
// ---------------------------------------------------------------------------
__global__ __launch_bounds__(256)
void gate_k(float* __restrict__ ys, const float* __restrict__ xc,
            const float* __restrict__ Dp, const float* __restrict__ xz)
{
    const int idx = blockIdx.x * blockDim.x + threadIdx.x;
    if (idx >= NTOK * D_INNER) return;
    const int d   = idx & (D_INNER - 1);
    const int tok = idx >> 10;
    const float y = ys[idx] + xc[idx] * Dp[d];
    const float z = xz[(size_t)tok * (2 * D_INNER) + D_INNER + d];
    ys[idx] = y * z * sigmoid_f(z);
}

// ---------------------------------------------------------------------------
// Host-side Mamba block launcher
// ---------------------------------------------------------------------------
static void run_mamba(const float* in, const float* res, float* out,
                      const float* Win, const float* convw, const float* convb,
                      const float* Wx, const float* Wdt, const float* bdt,
                      const float* Alog, const float* Dp, const float* Wout,
                      const float* ln_g, const float* ln_b,
                      float* lnx, float* xz, float* xc, float* xdbl,
                      float* dtb, float* ys, hipStream_t stream)
{
    const int NXD = DT_RANK + 2 * D_STATE;  // 64

    layernorm_k<<<NTOK / 8, 256, 0, stream>>>(in, ln_g, ln_b, lnx, NTOK);

    // xz = lnx @ Win : 4096 x 2048, K=512
    gemm_wmma_f32<0, 0, 0><<<dim3(NTOK / 128, (2 * D_INNER) / 64), 256, 0, stream>>>(
        lnx, Win, nullptr, nullptr, xz, NTOK, 2 * D_INNER, D_MODEL, D_MODEL);

    conv_silu_k<<<(NTOK * D_INNER) / 256, 256, 0, stream>>>(xz, convw, convb, xc);

    // xdbl = xc @ Wx : 4096 x 64, K=1024
    gemm_wmma_f32<0, 0, 0><<<dim3(NTOK / 128, NXD / 64), 256, 0, stream>>>(
        xc, Wx, nullptr, nullptr, xdbl, NTOK, NXD, D_INNER, D_INNER);

    // dt = softplus(xdbl[:, :32] @ Wdt + bdt) : 4096 x 1024, K=32, lda=64
    gemm_wmma_f32<1, 2, 0><<<dim3(NTOK / 128, D_INNER / 64), 256, 0, stream>>>(
        xdbl, Wdt, bdt, nullptr, dtb, NTOK, D_INNER, DT_RANK, NXD);

    scan_k<<<BATCH * (D_INNER / 256), 256, 0, stream>>>(dtb, xdbl, xc, Alog, ys);

    gate_k<<<(NTOK * D_INNER) / 256, 256, 0, stream>>>(ys, xc, Dp, xz);

    // out = res + ys @ Wout : 4096 x 512, K=1024
    gemm_wmma_f32<0, 0, 1><<<dim3(NTOK / 128, D_MODEL / 64), 256, 0, stream>>>(
        ys, Wout, nullptr, res, out, NTOK, D_MODEL, D_INNER, D_INNER);
}

// ---------------------------------------------------------------------------
// Entry point
// ---------------------------------------------------------------------------
extern "C" void kernel_launch(void* const* d_in, const int* in_sizes, int n_in,
                              void* d_out, int out_size, void* d_ws, size_t ws_size,
                              hipStream_t stream)
{
    (void)in_sizes; (void)n_in; (void)out_size; (void)ws_size;

    const float* x      = (const float*)d_in[0];
    const float* ln1_g  = (const float*)d_in[1];
    const float* ln1_b  = (const float*)d_in[2];
    const float* ln2_g  = (const float*)d_in[3];
    const float* ln2_b  = (const float*)d_in[4];
    const float* ln3_g  = (const float*)d_in[5];
    const float* ln3_b  = (const float*)d_in[6];
    const float* ffn_w1 = (const float*)d_in[7];
    const float* ffn_b1 = (const float*)d_in[8];
    const float* ffn_w2 = (const float*)d_in[9];
    const float* ffn_b2 = (const float*)d_in[10];
    // m1: 11..19, m2: 20..28 (Win, convw, convb, Wx, Wdt, bdt, Alog, D, Wout)
    const float* const* m1 = (const float* const*)&d_in[11];
    const float* const* m2 = (const float* const*)&d_in[20];

    float* out = (float*)d_out;

    // Workspace layout (floats)
    float* ws = (float*)d_ws;
    const size_t SZ_LNX  = (size_t)NTOK * D_MODEL;        // 2M
    const size_t SZ_XZ   = (size_t)NTOK * 2 * D_INNER;    // 8M
    const size_t SZ_XC   = (size_t)NTOK * D_INNER;        // 4M
    const size_t SZ_XDBL = (size_t)NTOK * 64;             // 256K
    float* lnx  = ws;
    float* xz   = lnx  + SZ_LNX;
    float* xc   = xz   + SZ_XZ;
    float* xdbl = xc   + SZ_XC;
    float* dtb  = xdbl + SZ_XDBL;
    float* ys   = dtb  + SZ_XC;
    float* hbuf = ys   + SZ_XC;                           // +2M -> ~97 MB total

    // h = x + mamba1(LN1(x))
    run_mamba(x, x, hbuf,
              m1[0], m1[1], m1[2], m1[3], m1[4], m1[5], m1[6], m1[7], m1[8],
              ln1_g, ln1_b, lnx, xz, xc, xdbl, dtb, ys, stream);

    // h = h + mamba2(LN2(h))   (residual GEMM reads/writes hbuf element-wise)
    run_mamba(hbuf, hbuf, hbuf,
              m2[0], m2[1], m2[2], m2[3], m2[4], m2[5], m2[6], m2[7], m2[8],
              ln2_g, ln2_b, lnx, xz, xc, xdbl, dtb, ys, stream);

    // FFN: f = LN3(h); g = gelu(f @ w1 + b1); out = h + g @ w2 + b2
    layernorm_k<<<NTOK / 8, 256, 0, stream>>>(hbuf, ln3_g, ln3_b, lnx, NTOK);

    gemm_wmma_f32<1, 1, 0><<<dim3(NTOK / 128, (4 * D_MODEL) / 64), 256, 0, stream>>>(
        lnx, ffn_w1, ffn_b1, nullptr, xz, NTOK, 4 * D_MODEL, D_MODEL, D_MODEL);

    gemm_wmma_f32<1, 0, 1><<<dim3(NTOK / 128, D_MODEL / 64), 256, 0, stream>>>(
        xz, ffn_w2, ffn_b2, hbuf, out, NTOK, D_MODEL, 4 * D_MODEL, 4 * D_MODEL);
}